// ConcatAttention_13709535609366
// MI455X (gfx1250) — compile-verified
//
#include <hip/hip_runtime.h>
#include <hip/hip_bf16.h>

// ---------------------------------------------------------------------------
// ConcatAttention (additive / MLP attention), MI455X gfx1250, wave32.
//   B=4, SQ=SK=256, DQ=DK=DV=256, D(hidden)=512.
// Pipeline:
//   [WMMA f32 16x16x4]  qh  = Q @ W1[:, :256]^T               -> ws
//   [WMMA f32 16x16x4]  khb = K @ W1[:, 256:]^T + b1          -> ws
//   [VALU + LDS + prefetch] scores = sum_h lrelu(qh+khb)*W2   -> ws
//   [LDS reduce]        softmax rows (in place)
//   [WMMA f32 16x16x4]  out = attn @ V (batched)
// GEMM: 32x32 tile per wave (4 accumulators, 2x2 fragment reuse), 4 waves
// per block (64x64), software-pipelined K loop so loads overlap the WMMA
// chain. All dims divide tile sizes exactly -> EXEC all-ones at every WMMA.
// Prefetches use locality 3 -> WGP-scope (fills near cache; SYS-scope
// prefetch would stop at GL2, useless for an already-L2-resident stream).
// ---------------------------------------------------------------------------

typedef float v2f __attribute__((ext_vector_type(2)));
typedef float v8f __attribute__((ext_vector_type(8)));

#define D_HID 512
#define SEQ   256
#define NB    4

#define WMMA_F32(A_, B_, C_) \
    __builtin_amdgcn_wmma_f32_16x16x4_f32(false, (A_), false, (B_), (short)0, (C_), false, false)

// ---------------------------------------------------------------------------
// Generic f32 WMMA GEMM:  C[m,n] = sum_k A[m*lda + k] * Bm[k*bs_k + n*bs_n]
//                                  (+ bias[n] if bias != nullptr)
// Grid: (M/64, N/64, batches), 128 threads (4 waves in 2x2 of 32x32 tiles).
// ---------------------------------------------------------------------------
__global__ __launch_bounds__(128)
void wmma_gemm_f32(const float* __restrict__ A,
                   const float* __restrict__ Bm,
                   float* __restrict__ C,
                   const float* __restrict__ bias,
                   int K, int lda, int bs_k, int bs_n, int ldc,
                   long long strideA, long long strideB, long long strideC)
{
    const int wave = threadIdx.x >> 5;   // 0..3
    const int lane = threadIdx.x & 31;
    const int tileM = blockIdx.x * 64 + (wave >> 1) * 32;   // wave's 32-row base
    const int tileN = blockIdx.y * 64 + (wave & 1) * 32;    // wave's 32-col base
    const int bz = blockIdx.z;

    A  += (long long)bz * strideA;
    Bm += (long long)bz * strideB;
    C  += (long long)bz * strideC;

    const int half = lane >> 4;          // 0: lanes 0-15, 1: lanes 16-31
    const int l16  = lane & 15;
    const int kb   = half * 2;           // K sub-offset per ISA A/B layout

    // Per-lane streaming pointers (advance by 4 K-steps each iteration).
    const float* pa0 = A  + (long long)(tileM + l16)      * lda + kb;
    const float* pa1 = A  + (long long)(tileM + l16 + 16) * lda + kb;
    const float* pb0 = Bm + (long long)kb * bs_k + (long long)(tileN + l16)      * bs_n;
    const float* pb1 = Bm + (long long)kb * bs_k + (long long)(tileN + l16 + 16) * bs_n;
    const long long bstep = 4LL * bs_k;

    v8f acc00 = {}, acc01 = {}, acc10 = {}, acc11 = {};

    // Prologue: fragments for k0 = 0.
    v2f a0, a1, b0, b1;
    a0.x = pa0[0]; a0.y = pa0[1];
    a1.x = pa1[0]; a1.y = pa1[1];
    b0.x = pb0[0]; b0.y = pb0[bs_k];
    b1.x = pb1[0]; b1.y = pb1[bs_k];

    // Software-pipelined main loop: load step i+1 fragments, then 4 WMMAs on i.
    for (int k0 = 0; k0 < K - 4; k0 += 4) {
        pa0 += 4; pa1 += 4; pb0 += bstep; pb1 += bstep;
        v2f na0, na1, nb0, nb1;
        na0.x = pa0[0]; na0.y = pa0[1];
        na1.x = pa1[0]; na1.y = pa1[1];
        nb0.x = pb0[0]; nb0.y = pb0[bs_k];
        nb1.x = pb1[0]; nb1.y = pb1[bs_k];
        __builtin_prefetch(pa0 + 16, 0, 3);   // WGP-scope, pull into near cache

        acc00 = WMMA_F32(a0, b0, acc00);
        acc01 = WMMA_F32(a0, b1, acc01);
        acc10 = WMMA_F32(a1, b0, acc10);
        acc11 = WMMA_F32(a1, b1, acc11);

        a0 = na0; a1 = na1; b0 = nb0; b1 = nb1;
    }
    // Epilogue: last K-step.
    acc00 = WMMA_F32(a0, b0, acc00);
    acc01 = WMMA_F32(a0, b1, acc01);
    acc10 = WMMA_F32(a1, b0, acc10);
    acc11 = WMMA_F32(a1, b1, acc11);

    // C/D layout: VGPR r of a 16x16 tile -> (M = r + half*8, N = l16).
    const int orow = tileM + half * 8;
    const int ocol = tileN + l16;
    const float bv0 = bias ? bias[ocol]      : 0.0f;
    const float bv1 = bias ? bias[ocol + 16] : 0.0f;
#pragma unroll
    for (int r = 0; r < 8; ++r) {
        C[(long long)(orow + r)      * ldc + ocol]      = acc00[r] + bv0;
        C[(long long)(orow + r)      * ldc + ocol + 16] = acc01[r] + bv1;
        C[(long long)(orow + r + 16) * ldc + ocol]      = acc10[r] + bv0;
        C[(long long)(orow + r + 16) * ldc + ocol + 16] = acc11[r] + bv1;
    }
}

// ---------------------------------------------------------------------------
// scores[b, qtile*16+q, k] = sum_h lrelu(qh[b,q',h] + khb[b,k,h]) * W2[h]
// Nonlinearity blocks matmul factoring -> VALU. qh tile + W2 in LDS (34KB),
// each thread streams one khb row (float4 + prefetch), amortized over 16 q.
// Grid: 64 blocks (b * qtile), 256 threads (one per k).
// ---------------------------------------------------------------------------
__global__ __launch_bounds__(256)
void scores_kernel(const float* __restrict__ qh,    // [B*SQ, 512]
                   const float* __restrict__ khb,   // [B*SK, 512] (bias folded)
                   const unsigned char* __restrict__ mask, // [B*SK] bool
                   float* __restrict__ scores,      // [B, SQ, SK]
                   const float* __restrict__ W2)    // [512]
{
    __shared__ float qh_s[16][D_HID];  // 32 KB
    __shared__ float w2_s[D_HID];      //  2 KB

    const int b     = blockIdx.x >> 4;
    const int qtile = blockIdx.x & 15;
    const int t     = threadIdx.x;

    const float* qbase = qh + (long long)(b * SEQ + qtile * 16) * D_HID;
    for (int i = t; i < 16 * D_HID; i += 256)
        qh_s[i >> 9][i & (D_HID - 1)] = qbase[i];
    for (int i = t; i < D_HID; i += 256)
        w2_s[i] = W2[i];
    __syncthreads();

    const int k = t;
    const float* krow = khb + (long long)(b * SEQ + k) * D_HID;

    float acc[16];
#pragma unroll
    for (int q = 0; q < 16; ++q) acc[q] = 0.0f;

    for (int h0 = 0; h0 < D_HID; h0 += 32) {
        // Prefetch next 128B chunk of this thread's khb row into the WGP
        // cache (speculative; OOB at the row end is silently dropped).
        __builtin_prefetch(krow + h0 + 32, 0, 3);
        for (int hh = 0; hh < 32; hh += 4) {
            const int h = h0 + hh;
            const float4 kv = *(const float4*)(krow + h);
            const float w0 = w2_s[h + 0], w1 = w2_s[h + 1];
            const float w2v = w2_s[h + 2], w3 = w2_s[h + 3];
#pragma unroll
            for (int q = 0; q < 16; ++q) {
                float s;
                s = qh_s[q][h + 0] + kv.x; acc[q] += fmaxf(s, 0.01f * s) * w0;
                s = qh_s[q][h + 1] + kv.y; acc[q] += fmaxf(s, 0.01f * s) * w1;
                s = qh_s[q][h + 2] + kv.z; acc[q] += fmaxf(s, 0.01f * s) * w2v;
                s = qh_s[q][h + 3] + kv.w; acc[q] += fmaxf(s, 0.01f * s) * w3;
            }
        }
    }

    const bool m = mask[b * SEQ + k] != 0;
#pragma unroll
    for (int q = 0; q < 16; ++q) {
        const long long row = (long long)(b * SEQ + qtile * 16 + q);
        scores[row * SEQ + k] = m ? -1e9f : acc[q];
    }
}

// ---------------------------------------------------------------------------
// In-place row softmax over SK=256. One 256-thread block per (b,q) row.
// ---------------------------------------------------------------------------
__global__ __launch_bounds__(256)
void softmax_kernel(float* __restrict__ scores)
{
    __shared__ float red[256];
    const long long row = blockIdx.x;
    const int t = threadIdx.x;

    const float v = scores[row * SEQ + t];

    red[t] = v;
    __syncthreads();
    for (int s = 128; s > 0; s >>= 1) {
        if (t < s) red[t] = fmaxf(red[t], red[t + s]);
        __syncthreads();
    }
    const float mx = red[0];
    __syncthreads();

    const float e = __expf(v - mx);
    red[t] = e;
    __syncthreads();
    for (int s = 128; s > 0; s >>= 1) {
        if (t < s) red[t] += red[t + s];
        __syncthreads();
    }
    scores[row * SEQ + t] = e * (1.0f / red[0]);
}

// ---------------------------------------------------------------------------
// Launch. Inputs (setup_inputs order): Q, K, V, mask_out, W1, b1, W2.
// Workspace: qh (2MB) | khb (2MB) | scores (1MB).
// ---------------------------------------------------------------------------
extern "C" void kernel_launch(void* const* d_in, const int* in_sizes, int n_in,
                              void* d_out, int out_size, void* d_ws, size_t ws_size,
                              hipStream_t stream)
{
    const float* Q  = (const float*)d_in[0];
    const float* Km = (const float*)d_in[1];
    const float* V  = (const float*)d_in[2];
    const unsigned char* mask = (const unsigned char*)d_in[3];
    const float* W1 = (const float*)d_in[4];
    const float* b1 = (const float*)d_in[5];
    const float* W2 = (const float*)d_in[6];
    float* out = (float*)d_out;

    float* qh  = (float*)d_ws;                       // [1024, 512]
    float* khb = qh + (size_t)NB * SEQ * D_HID;      // [1024, 512]
    float* sc  = khb + (size_t)NB * SEQ * D_HID;     // [4, 256, 256]

    const dim3 blk(128);

    // qh = Q @ W1[:, :256]^T : Bm[k][n] = W1[n*512 + k]  -> bs_k=1, bs_n=512
    wmma_gemm_f32<<<dim3((NB * SEQ) / 64, D_HID / 64, 1), blk, 0, stream>>>(
        Q, W1, qh, nullptr,
        /*K=*/256, /*lda=*/256, /*bs_k=*/1, /*bs_n=*/512, /*ldc=*/D_HID,
        0, 0, 0);

    // khb = K @ W1[:, 256:]^T + b1 : Bm[k][n] = W1[n*512 + 256 + k]
    wmma_gemm_f32<<<dim3((NB * SEQ) / 64, D_HID / 64, 1), blk, 0, stream>>>(
        Km, W1 + 256, khb, b1,
        /*K=*/256, /*lda=*/256, /*bs_k=*/1, /*bs_n=*/512, /*ldc=*/D_HID,
        0, 0, 0);

    // scores + mask (leaky-relu MLP scoring, VALU)
    scores_kernel<<<NB * (SEQ / 16), 256, 0, stream>>>(qh, khb, mask, sc, W2);

    // softmax rows
    softmax_kernel<<<NB * SEQ, 256, 0, stream>>>(sc);

    // out = attn @ V, batched: Bm[k][n] = V[k*256 + n] -> bs_k=256, bs_n=1
    wmma_gemm_f32<<<dim3(SEQ / 64, SEQ / 64, NB), blk, 0, stream>>>(
        sc, V, out, nullptr,
        /*K=*/256, /*lda=*/256, /*bs_k=*/256, /*bs_n=*/1, /*ldc=*/256,
        (long long)SEQ * SEQ, (long long)SEQ * SEQ, (long long)SEQ * SEQ);
}